// MaskedObjectPooling_61400852464179
// MI455X (gfx1250) — compile-verified
//
#include <hip/hip_runtime.h>
#include <hip/hip_bf16.h>
#include <climits>

// ---------------- problem constants (match reference) ----------------
constexpr int BATCH   = 8;
constexpr int KOBJ    = 32;
constexpr int HH      = 480;
constexpr int WW      = 480;
constexpr int NTOK    = 576;    // 24*24
constexpr int DDIM    = 1024;
constexpr int CLS_DIM = 64;
constexpr int OUT     = 256;
constexpr int IN_DIM  = DDIM + CLS_DIM + 5; // 1093
constexpr int IN_PAD  = 1096;               // round K up to multiple of 4
constexpr int BK      = BATCH * KOBJ;       // 256 rows through the MLP

typedef float v2f __attribute__((ext_vector_type(2)));
typedef float v8f __attribute__((ext_vector_type(8)));

// ---------------- wave reductions (wave32) ----------------
__device__ __forceinline__ int wredAdd(int v){ for (int o=16;o>0;o>>=1) v += __shfl_xor(v,o,32); return v; }
__device__ __forceinline__ int wredMin(int v){ for (int o=16;o>0;o>>=1) v = min(v,__shfl_xor(v,o,32)); return v; }
__device__ __forceinline__ int wredMax(int v){ for (int o=16;o>0;o>>=1) v = max(v,__shfl_xor(v,o,32)); return v; }

// ---------------- 0) init per-object stats ----------------
// stats[bk][0..7] = area, sum_x, sum_y, xmin, xmax, ymin, ymax, pad
__global__ void init_stats_kernel(int* __restrict__ stats) {
    int i = blockIdx.x * blockDim.x + threadIdx.x;
    if (i < BK * 8) {
        int f = i & 7;
        int v = 0;
        if (f == 3 || f == 5) v = INT_MAX;
        if (f == 4 || f == 6) v = INT_MIN;
        stats[i] = v;
    }
}

// ---------------- 1) streaming mask-stats pass (the HBM-bound kernel) --------
// grid = (BK, CHUNKS); int4 loads; wave-shuffle reduce; 7 atomics per wave.
constexpr int CHUNKS = 8;
__global__ void mask_stats_kernel(const int* __restrict__ masks, int* __restrict__ stats) {
    const int bk      = blockIdx.x;
    const int quads   = (HH * WW) / 4;         // 57600 int4 per mask
    const int perChnk = quads / CHUNKS;        // 7200
    const int start   = blockIdx.y * perChnk;
    const int4* mp    = (const int4*)(masks + (size_t)bk * HH * WW);

    int area = 0, sx = 0, sy = 0;
    int xmin = INT_MAX, xmax = INT_MIN, ymin = INT_MAX, ymax = INT_MIN;

    for (int i = start + threadIdx.x; i < start + perChnk; i += blockDim.x) {
        __builtin_prefetch(mp + i + blockDim.x, 0, 0);   // global_prefetch_b8
        int4 v = mp[i];
        int p  = i * 4;
        int y  = p / WW;
        int x0 = p - y * WW;                   // W%4==0: an int4 never crosses rows
        int m0 = v.x > 0, m1 = v.y > 0, m2 = v.z > 0, m3 = v.w > 0;
        int cnt = m0 + m1 + m2 + m3;
        area += cnt;
        sx   += m0 * x0 + m1 * (x0 + 1) + m2 * (x0 + 2) + m3 * (x0 + 3);
        sy   += cnt * y;
        if (cnt) {
            ymin = min(ymin, y); ymax = max(ymax, y);
            int lx = m0 ? x0     : (m1 ? x0 + 1 : (m2 ? x0 + 2 : x0 + 3));
            int hx = m3 ? x0 + 3 : (m2 ? x0 + 2 : (m1 ? x0 + 1 : x0));
            xmin = min(xmin, lx); xmax = max(xmax, hx);
        }
    }
    area = wredAdd(area); sx = wredAdd(sx); sy = wredAdd(sy);
    xmin = wredMin(xmin); ymin = wredMin(ymin);
    xmax = wredMax(xmax); ymax = wredMax(ymax);
    if ((threadIdx.x & 31) == 0) {
        int* s = stats + bk * 8;
        atomicAdd(&s[0], area); atomicAdd(&s[1], sx); atomicAdd(&s[2], sy);
        atomicMin(&s[3], xmin); atomicMax(&s[4], xmax);
        atomicMin(&s[5], ymin); atomicMax(&s[6], ymax);
    }
}

// ---------------- 2) nearest-neighbor downsample + msum ----------------
__global__ void downsample_kernel(const int* __restrict__ masks,
                                  float* __restrict__ m_flat,
                                  float* __restrict__ msum) {
    int bk = blockIdx.x, t = threadIdx.x;
    const int* mp = masks + (size_t)bk * HH * WW;
    float acc = 0.f;
    for (int n = t; n < NTOK; n += blockDim.x) {
        int hf = n / 24, wf = n - hf * 24;      // idx = floor(dst*in/out) = dst*20
        float v = (mp[(hf * 20) * WW + wf * 20] > 0) ? 1.f : 0.f;
        m_flat[(size_t)bk * NTOK + n] = v;
        acc += v;
    }
    __shared__ float red[256];
    red[t] = acc; __syncthreads();
    for (int s = 128; s > 0; s >>= 1) { if (t < s) red[t] += red[t + s]; __syncthreads(); }
    if (t == 0) msum[bk] = fmaxf(red[0], 1e-6f);
}

// ---------------- 3) zero-pad W1 rows 1093..1095 ----------------
__global__ void pad_w1_kernel(const float* __restrict__ W1, float* __restrict__ w1pad) {
    int i = blockIdx.x * blockDim.x + threadIdx.x;
    if (i < IN_PAD * OUT) {
        int kk = i / OUT;
        w1pad[i] = (kk < IN_DIM) ? W1[i] : 0.f;
    }
}

// ---------------- shared f32 WMMA 16x16 tile (K loop, step 4) ----------------
// A: 16 x kdim (row-major, lda); B: kdim x 16 (row-major, ldb). Fragment layouts
// per CDNA5 ISA 7.12.2: A lanes 0-15 hold K=0,1; lanes 16-31 hold K=2,3.
__device__ __forceinline__ v8f wmma_tile_f32(const float* __restrict__ A, int lda,
                                             const float* __restrict__ Bm, int ldb,
                                             int kdim, int lane) {
    v8f acc = {0.f,0.f,0.f,0.f,0.f,0.f,0.f,0.f};
    const int h2 = (lane >> 4) * 2;
    const float* arow = A + (size_t)(lane & 15) * lda + h2;
    const float* bcol = Bm + (lane & 15) + (size_t)h2 * ldb;
    for (int kk = 0; kk < kdim; kk += 4) {
        float2 a2 = *(const float2*)(arow + kk);
        v2f a = { a2.x, a2.y };
        v2f b = { bcol[(size_t)kk * ldb], bcol[(size_t)(kk + 1) * ldb] };
        acc = __builtin_amdgcn_wmma_f32_16x16x4_f32(false, a, false, b,
                                                    (short)0, acc, false, false);
    }
    return acc;
}

// ---------------- 4) pooled[b] = m_flat[b] (32x576) @ feature_map[b] (576x1024)
// 1 wave per 16x16 output tile: 8 batches * 2 * 64 tiles = 1024 waves.
__global__ void pooled_gemm_kernel(const float* __restrict__ m_flat,
                                   const float* __restrict__ fmap,
                                   float* __restrict__ pooled) {
    int w    = blockIdx.x * (blockDim.x >> 5) + (threadIdx.x >> 5);
    int lane = threadIdx.x & 31;
    int b    = w >> 7;
    int rem  = w & 127;
    int mt   = rem >> 6;     // 0..1
    int nt   = rem & 63;     // 0..63
    const float* A  = m_flat + ((size_t)b * KOBJ + mt * 16) * NTOK;
    const float* Bm = fmap   + (size_t)b * NTOK * DDIM + nt * 16;
    v8f acc = wmma_tile_f32(A, NTOK, Bm, DDIM, NTOK, lane);
    int col = nt * 16 + (lane & 15);
    int mh  = 8 * (lane >> 4);
    #pragma unroll
    for (int v = 0; v < 8; ++v) {
        int m = mt * 16 + v + mh;
        pooled[((size_t)b * KOBJ + m) * DDIM + col] = acc[v];
    }
}

// ---------------- 5) assemble x = [pooled/msum | cls_emb | geom | 0pad] -------
__global__ void build_x_kernel(const float* __restrict__ pooled,
                               const float* __restrict__ msum,
                               const int*   __restrict__ stats,
                               const int*   __restrict__ class_ids,
                               const float* __restrict__ embed,
                               float* __restrict__ xmat) {
    int bk = blockIdx.x;
    const int* s = stats + bk * 8;
    float areaF = (float)s[0];
    float safe  = fmaxf(areaF, 1.0f);
    bool  valid = areaF >= 1.0f;
    float geom[5];
    geom[0] = valid ? (float)s[1] / safe / (float)WW : 0.f;                 // cx
    geom[1] = valid ? (float)s[2] / safe / (float)HH : 0.f;                 // cy
    geom[2] = valid ? areaF / (float)(HH * WW) : 0.f;                       // area_norm
    geom[3] = valid ? (float)(s[4] - s[3] + 1) / (float)WW : 0.f;           // bbox_w
    geom[4] = valid ? (float)(s[6] - s[5] + 1) / (float)HH : 0.f;           // bbox_h
    int   cid = class_ids[bk];
    float inv = 1.0f / msum[bk];
    for (int c = threadIdx.x; c < IN_PAD; c += blockDim.x) {
        float v;
        if      (c < DDIM)           v = pooled[(size_t)bk * DDIM + c] * inv;
        else if (c < DDIM + CLS_DIM) v = embed[cid * CLS_DIM + (c - DDIM)];
        else if (c < IN_DIM)         v = geom[c - (DDIM + CLS_DIM)];
        else                         v = 0.f;
        xmat[(size_t)bk * IN_PAD + c] = v;
    }
}

// ---------------- 6) h = gelu_exact(x @ W1 + b1) : 256x1096 @ 1096x256 --------
__global__ void gemm1_kernel(const float* __restrict__ xmat,
                             const float* __restrict__ w1pad,
                             const float* __restrict__ b1,
                             float* __restrict__ hmat) {
    int w    = blockIdx.x * 8 + (threadIdx.x >> 5);
    int lane = threadIdx.x & 31;
    int mt   = w >> 4, nt = w & 15;
    v8f acc = wmma_tile_f32(xmat + (size_t)mt * 16 * IN_PAD, IN_PAD,
                            w1pad + nt * 16, OUT, IN_PAD, lane);
    int col = nt * 16 + (lane & 15);
    float bb = b1[col];
    int  mh  = 8 * (lane >> 4);
    #pragma unroll
    for (int v = 0; v < 8; ++v) {
        int m = mt * 16 + v + mh;
        float val = acc[v] + bb;
        val = 0.5f * val * (1.0f + erff(val * 0.70710678118654752440f)); // exact GELU
        hmat[(size_t)m * OUT + col] = val;
    }
}

// ---------------- 7) LayerNorm over OUT=256 cols, one block per row ----------
__global__ void ln_kernel(const float* __restrict__ hmat,
                          const float* __restrict__ g,
                          const float* __restrict__ b,
                          float* __restrict__ hn) {
    int row = blockIdx.x, t = threadIdx.x;
    __shared__ float red[256];
    float v = hmat[(size_t)row * OUT + t];
    red[t] = v; __syncthreads();
    for (int s = 128; s > 0; s >>= 1) { if (t < s) red[t] += red[t + s]; __syncthreads(); }
    float mu = red[0] * (1.0f / OUT);
    __syncthreads();
    float d = v - mu;
    red[t] = d * d; __syncthreads();
    for (int s = 128; s > 0; s >>= 1) { if (t < s) red[t] += red[t + s]; __syncthreads(); }
    float var = red[0] * (1.0f / OUT);
    hn[(size_t)row * OUT + t] = d * rsqrtf(var + 1e-5f) * g[t] + b[t];
}

// ---------------- 8) out = hn @ W2 + b2 : 256x256 @ 256x256 ------------------
__global__ void gemm2_kernel(const float* __restrict__ hn,
                             const float* __restrict__ W2,
                             const float* __restrict__ b2,
                             float* __restrict__ out) {
    int w    = blockIdx.x * 8 + (threadIdx.x >> 5);
    int lane = threadIdx.x & 31;
    int mt   = w >> 4, nt = w & 15;
    v8f acc = wmma_tile_f32(hn + (size_t)mt * 16 * OUT, OUT,
                            W2 + nt * 16, OUT, OUT, lane);
    int col = nt * 16 + (lane & 15);
    float bb = b2[col];
    int  mh  = 8 * (lane >> 4);
    #pragma unroll
    for (int v = 0; v < 8; ++v) {
        int m = mt * 16 + v + mh;
        out[(size_t)m * OUT + col] = acc[v] + bb;
    }
}

// ---------------- host side ----------------
extern "C" void kernel_launch(void* const* d_in, const int* in_sizes, int n_in,
                              void* d_out, int out_size, void* d_ws, size_t ws_size,
                              hipStream_t stream) {
    const float* fmap      = (const float*)d_in[0];
    const int*   masks     = (const int*)  d_in[1];
    const int*   class_ids = (const int*)  d_in[2];
    const float* embed     = (const float*)d_in[3];
    const float* W1        = (const float*)d_in[4];
    const float* b1        = (const float*)d_in[5];
    const float* ln_g      = (const float*)d_in[6];
    const float* ln_b      = (const float*)d_in[7];
    const float* W2        = (const float*)d_in[8];
    const float* b2        = (const float*)d_in[9];
    float*       out       = (float*)d_out;
    (void)in_sizes; (void)n_in; (void)out_size; (void)ws_size;

    char* ws = (char*)d_ws;
    size_t off = 0;
    auto carve = [&](size_t bytes) -> char* {
        char* p = ws + off;
        off = (off + bytes + 255) & ~(size_t)255;
        return p;
    };
    int*   stats  = (int*)  carve((size_t)BK * 8 * sizeof(int));
    float* m_flat = (float*)carve((size_t)BK * NTOK * sizeof(float));
    float* msum   = (float*)carve((size_t)BK * sizeof(float));
    float* pooled = (float*)carve((size_t)BK * DDIM * sizeof(float));
    float* xmat   = (float*)carve((size_t)BK * IN_PAD * sizeof(float));
    float* w1pad  = (float*)carve((size_t)IN_PAD * OUT * sizeof(float));
    float* hmat   = (float*)carve((size_t)BK * OUT * sizeof(float));
    float* hn     = (float*)carve((size_t)BK * OUT * sizeof(float));

    init_stats_kernel<<<(BK * 8 + 255) / 256, 256, 0, stream>>>(stats);
    mask_stats_kernel<<<dim3(BK, CHUNKS), 256, 0, stream>>>(masks, stats);
    downsample_kernel<<<BK, 256, 0, stream>>>(masks, m_flat, msum);
    pad_w1_kernel<<<(IN_PAD * OUT + 255) / 256, 256, 0, stream>>>(W1, w1pad);
    pooled_gemm_kernel<<<128, 256, 0, stream>>>(m_flat, fmap, pooled);
    build_x_kernel<<<BK, 256, 0, stream>>>(pooled, msum, stats, class_ids, embed, xmat);
    gemm1_kernel<<<32, 256, 0, stream>>>(xmat, w1pad, b1, hmat);
    ln_kernel<<<BK, 256, 0, stream>>>(hmat, ln_g, ln_b, hn);
    gemm2_kernel<<<32, 256, 0, stream>>>(hn, W2, b2, out);
}